// SingleQubitReuploadCell_17875653886661
// MI455X (gfx1250) — compile-verified
//
#include <hip/hip_runtime.h>
#include <cstdint>

#define BLOCK 256
#define WLEN 64
#define CHUNK 16
#define NCHUNK (WLEN / CHUNK)        // 4 chunks of 16 steps
#define QPR (CHUNK / 4)              // 4 float4 per row per chunk
#define STRIDE 20                    // LDS row stride in floats: 80B (16B-aligned, <=2-way bank conflict)
#define TILE_ELEMS (BLOCK * STRIDE)  // 5120 floats per buffer

#define INV2PI 0.15915494309189535f  // v_sin/v_cos take revolutions

__global__ __launch_bounds__(BLOCK) void qreupload_kernel(
    const float* __restrict__ x, const float* __restrict__ theta,
    const float* __restrict__ w, const float* __restrict__ bias,
    float* __restrict__ out, int nrows)
{
    __shared__ float  tile[2][TILE_ELEMS];   // double-buffered x tiles (async-filled)
    __shared__ float4 trig[WLEN];            // (cy, sy, cos th1, sin th1) per step

    const int tid     = threadIdx.x;
    const int rowBase = blockIdx.x * BLOCK;
    const int row     = rowBase + tid;

    // Per-block theta trig table. theta ~ 0.01*N(0,1): hardware v_sin/v_cos ample.
    // RZ(th1) == e^{-i th1/2} * diag(1, e^{+i th1}); the global phase cannot affect
    // <Z>, so we store full-angle (cos th1, sin th1) and rotate only s1 (4 ops).
    if (tid < WLEN) {
        float h0 = theta[2 * tid + 0] * (0.5f * INV2PI);   // half-angle, revolutions
        float h1 = theta[2 * tid + 1] * INV2PI;            // FULL angle, revolutions
        float cy = __builtin_amdgcn_cosf(h0);
        float sy = __builtin_amdgcn_sinf(h0);
        float c1 = __builtin_amdgcn_cosf(h1);
        float s1 = __builtin_amdgcn_sinf(h1);
        trig[tid] = make_float4(cy, sy, c1, s1);
    }

    // Issue one chunk of async global->LDS b128 copies (4 per thread, coalesced:
    // consecutive lanes cover consecutive 16B quads within 64B row segments).
    auto issue = [&](int ch, int buf) {
#pragma unroll
        for (int k = 0; k < QPR; ++k) {
            int m = k * BLOCK + tid;
            int r = m >> 2;            // tile row 0..255
            int q = m & 3;             // quad within chunk
            int grow = rowBase + r;
            if (grow > nrows - 1) grow = nrows - 1;   // safety clamp (B divides exactly)
            const float* gp = x + (size_t)grow * WLEN + ch * CHUNK + q * 4;
            unsigned lds32 = (unsigned)(uintptr_t)(&tile[buf][r * STRIDE + q * 4]);
            asm volatile("global_load_async_to_lds_b128 %0, %1, off"
                         :: "v"(lds32), "v"((unsigned long long)(uintptr_t)gp)
                         : "memory");
        }
    };

    issue(0, 0);

    // state |psi> = (a0 + i b0, a1 + i b1), start |0> (up to dropped global phase)
    float a0 = 1.0f, b0 = 0.0f, a1 = 0.0f, b1 = 0.0f;

#pragma unroll
    for (int ch = 0; ch < NCHUNK; ++ch) {
        if (ch + 1 < NCHUNK) {
            issue(ch + 1, (ch + 1) & 1);
            // 8 async ops outstanding; oldest chunk (4 ops) completes in order
            asm volatile("s_wait_asynccnt 0x4" ::: "memory");
        } else {
            asm volatile("s_wait_asynccnt 0x0" ::: "memory");
        }
        __syncthreads();   // all waves' async fills (and trig table) visible

        const float* rowp = &tile[ch & 1][tid * STRIDE];
#pragma unroll
        for (int j4 = 0; j4 < CHUNK; j4 += 4) {
            float4 xq = *(const float4*)(rowp + j4);   // ds_load_b128
#pragma unroll
            for (int j = 0; j < 4; ++j) {
                float xt = (j == 0) ? xq.x : (j == 1) ? xq.y : (j == 2) ? xq.z : xq.w;
                float4 tg = trig[ch * CHUNK + j4 + j]; // broadcast ds_load_b128
                // half-angle in revolutions: one mul + 2 trans ops
                float k = xt * (0.5f * INV2PI);
                float s = __builtin_amdgcn_sinf(k);    // v_sin_f32
                float c = __builtin_amdgcn_cosf(k);    // v_cos_f32
                // RX(xt)
                float n0r = fmaf(c, a0,  s * b1);
                float n0i = fmaf(c, b0, -s * a1);
                float n1r = fmaf(c, a1,  s * b0);
                float n1i = fmaf(c, b1, -s * a0);
                // RY(th0): (cy = tg.x, sy = tg.y)
                float m0r = fmaf(tg.x, n0r, -tg.y * n1r);
                float m0i = fmaf(tg.x, n0i, -tg.y * n1i);
                float m1r = fmaf(tg.y, n0r,  tg.x * n1r);
                float m1i = fmaf(tg.y, n0i,  tg.x * n1i);
                // RZ(th1) modulo global phase: s0 = m0, s1 = e^{i th1} * m1
                a0 = m0r;
                b0 = m0i;
                a1 = fmaf(tg.z, m1r, -tg.w * m1i);
                b1 = fmaf(tg.z, m1i,  tg.w * m1r);
            }
        }
        __syncthreads();   // tile[ch&1] free for reuse before next overwrite
    }

    float z = (a0 * a0 + b0 * b0) - (a1 * a1 + b1 * b1);
    if (row < nrows) out[row] = fmaf(z, w[0], bias[0]);
}

extern "C" void kernel_launch(void* const* d_in, const int* in_sizes, int n_in,
                              void* d_out, int out_size, void* d_ws, size_t ws_size,
                              hipStream_t stream) {
    const float* x     = (const float*)d_in[0];
    const float* theta = (const float*)d_in[1];
    const float* w     = (const float*)d_in[2];
    const float* bias  = (const float*)d_in[3];
    float* out = (float*)d_out;
    int nrows = out_size;                      // B = 524288
    int grid  = (nrows + BLOCK - 1) / BLOCK;   // 2048 blocks of 8 wave32
    hipLaunchKernelGGL(qreupload_kernel, dim3(grid), dim3(BLOCK), 0, stream,
                       x, theta, w, bias, out, nrows);
}